// RGCNN_11553462026389
// MI455X (gfx1250) — compile-verified
//
#include <hip/hip_runtime.h>
#include <hip/hip_bf16.h>
#include <math.h>

typedef __attribute__((ext_vector_type(2))) float v2f;
typedef __attribute__((ext_vector_type(8))) float v8f;

// ---------------------------------------------------------------------------
// WMMA f32 GEMM: Y[z] = X @ W[z] (+ bias), tiles of 16x16, K stepped by 4.
// One wave (32 threads) per 16x16 tile. N multiple of 16, O multiple of 16,
// I multiple of 4 (holds: N=100000=16*6250, I in {16,32}, O in {16,32,64}).
// A 16x4 f32 layout (ISA 7.12.2): lane<16 -> row=lane, v0=K+0, v1=K+1;
//                                 lane>=16 -> row=lane-16, v0=K+2, v1=K+3.
// B 4x16 layout (row striped across lanes, symmetric): lane<16 -> col=lane,
//   v0 = W[K+0][col], v1 = W[K+1][col]; lane>=16 -> v0=W[K+2][col], v1=W[K+3][col].
// D 16x16: acc[j] -> row = m0 + j + 8*(lane>=16), col = n0 + (lane&15).
// ---------------------------------------------------------------------------
__global__ __launch_bounds__(32)
void gemm16_wmma_f32(const float* __restrict__ X, const float* __restrict__ Wall,
                     const float* __restrict__ bias, float* __restrict__ Yall,
                     int N, int I, int O, int wStride, int yStride) {
    const int m0 = blockIdx.x * 16;
    const int n0 = blockIdx.y * 16;
    const float* __restrict__ W = Wall + (size_t)blockIdx.z * wStride;
    float* __restrict__ Y = Yall + (size_t)blockIdx.z * yStride;

    const int lane = threadIdx.x;
    const int hi   = lane >> 4;    // 0 or 1
    const int l15  = lane & 15;

    v8f acc = {};
    for (int k = 0; k < I; k += 4) {
        v2f a, b;
        const float* xp = X + (size_t)(m0 + l15) * I + (k + 2 * hi);
        a.x = xp[0];
        a.y = xp[1];
        const float* wp = W + (size_t)(k + 2 * hi) * O + (n0 + l15);
        b.x = wp[0];
        b.y = wp[O];
        acc = __builtin_amdgcn_wmma_f32_16x16x4_f32(
            /*neg_a=*/false, a, /*neg_b=*/false, b,
            /*c_mod=*/(short)0, acc, /*reuse_a=*/false, /*reuse_b=*/false);
    }

    const float bval = bias ? bias[n0 + l15] : 0.0f;
    #pragma unroll
    for (int j = 0; j < 8; ++j) {
        const int row = m0 + j + hi * 8;
        Y[(size_t)row * O + (n0 + l15)] = acc[j] + bval;
    }
}

// ---------------------------------------------------------------------------
// Utility fills
// ---------------------------------------------------------------------------
__global__ void fill_i32(int* __restrict__ p, int v, int n) {
    int t = blockIdx.x * blockDim.x + threadIdx.x;
    if (t < n) p[t] = v;
}
__global__ void fill_f32(float* __restrict__ p, float v, int n) {
    int t = blockIdx.x * blockDim.x + threadIdx.x;
    if (t < n) p[t] = v;
}

// ---------------------------------------------------------------------------
// Count edges per (dst, relation) segment: cnt[dst*4 + etype]++
// ---------------------------------------------------------------------------
__global__ void count_edges(const int* __restrict__ dst, const int* __restrict__ et,
                            int* __restrict__ cnt, int E) {
    int t = blockIdx.x * blockDim.x + threadIdx.x;
    if (t < E) atomicAdd(&cnt[dst[t] * 4 + et[t]], 1);
}

// ---------------------------------------------------------------------------
// Scatter mean-aggregated messages: out[dst] += xr[etype][src] / max(cnt,1)
// One thread per (edge, out-channel). O is a power of two (oShift = log2 O).
// ---------------------------------------------------------------------------
__global__ void scatter_mean(const float* __restrict__ xr, const int* __restrict__ src,
                             const int* __restrict__ dst, const int* __restrict__ et,
                             const int* __restrict__ cnt, float* __restrict__ out,
                             long long total, int O, int N, int oShift) {
    long long t = (long long)blockIdx.x * blockDim.x + threadIdx.x;
    if (t >= total) return;
    int e = (int)(t >> oShift);
    int o = (int)(t & (O - 1));
    int d = dst[e];
    int r = et[e];
    float c = (float)cnt[d * 4 + r];
    float norm = 1.0f / fmaxf(c, 1.0f);
    float m = xr[((size_t)r * N + src[e]) * O + o] * norm;
    atomicAdd(&out[(size_t)d * O + o], m);
}

__global__ void relu_inplace(float* __restrict__ p, int n) {
    int t = blockIdx.x * blockDim.x + threadIdx.x;
    if (t < n) p[t] = fmaxf(p[t], 0.0f);
}

// ---------------------------------------------------------------------------
// Segment max pool: g[batch[n]][o] = max over nodes (sign-aware int atomics)
// ---------------------------------------------------------------------------
__device__ __forceinline__ void atomicMaxFloat(float* addr, float val) {
    if (val >= 0.0f) {
        atomicMax((int*)addr, __float_as_int(val));
    } else {
        atomicMin((unsigned int*)addr, __float_as_uint(val));
    }
}

__global__ void pool_max(const float* __restrict__ h, const int* __restrict__ batch,
                         float* __restrict__ g, int N) {
    int t = blockIdx.x * blockDim.x + threadIdx.x;
    if (t >= N * 64) return;
    int n = t >> 6;
    int o = t & 63;
    atomicMaxFloat(&g[batch[n] * 64 + o], h[(size_t)n * 64 + o]);
}

// ---------------------------------------------------------------------------
// MLP head: one block per row, input row staged in LDS
// ---------------------------------------------------------------------------
__global__ void linear_act(const float* __restrict__ A, const float* __restrict__ W,
                           const float* __restrict__ b, float* __restrict__ Y,
                           int IN, int OUT, int doRelu) {
    __shared__ float row[256];
    int r = blockIdx.x;
    for (int i = threadIdx.x; i < IN; i += blockDim.x) row[i] = A[(size_t)r * IN + i];
    __syncthreads();
    int o = threadIdx.x;
    if (o < OUT) {
        float s = b[o];
        for (int i = 0; i < IN; ++i) s += row[i] * W[(size_t)i * OUT + o];
        if (doRelu) s = fmaxf(s, 0.0f);
        Y[(size_t)r * OUT + o] = s;
    }
}

__global__ void final_linear_softmax(const float* __restrict__ A, const float* __restrict__ W,
                                     const float* __restrict__ b, float* __restrict__ out,
                                     int IN, int C) {
    __shared__ float row[256];
    __shared__ float logits[64];
    __shared__ float s_max, s_sum;
    int r = blockIdx.x;
    for (int i = threadIdx.x; i < IN; i += blockDim.x) row[i] = A[(size_t)r * IN + i];
    __syncthreads();
    int o = threadIdx.x;
    if (o < C) {
        float s = b[o];
        for (int i = 0; i < IN; ++i) s += row[i] * W[(size_t)i * C + o];
        logits[o] = s;
    }
    __syncthreads();
    if (threadIdx.x == 0) {
        float m = -INFINITY;
        for (int i = 0; i < C; ++i) m = fmaxf(m, logits[i]);
        float s = 0.0f;
        for (int i = 0; i < C; ++i) s += expf(logits[i] - m);
        s_max = m;
        s_sum = s;
    }
    __syncthreads();
    if (o < C) out[(size_t)r * C + o] = expf(logits[o] - s_max) / s_sum;
}

// ---------------------------------------------------------------------------
// Launch
// ---------------------------------------------------------------------------
extern "C" void kernel_launch(void* const* d_in, const int* in_sizes, int n_in,
                              void* d_out, int out_size, void* d_ws, size_t ws_size,
                              hipStream_t stream) {
    const float* x     = (const float*)d_in[0];
    const int*   ei    = (const int*)d_in[1];
    const int*   et    = (const int*)d_in[2];
    const int*   batch = (const int*)d_in[3];
    const float* W1    = (const float*)d_in[4];
    const float* root1 = (const float*)d_in[5];
    const float* b1    = (const float*)d_in[6];
    const float* W2    = (const float*)d_in[7];
    const float* root2 = (const float*)d_in[8];
    const float* b2    = (const float*)d_in[9];
    const float* W3    = (const float*)d_in[10];
    const float* root3 = (const float*)d_in[11];
    const float* b3    = (const float*)d_in[12];
    const float* Wf1   = (const float*)d_in[13];
    const float* bf1   = (const float*)d_in[14];
    const float* Wh    = (const float*)d_in[15];
    const float* bh    = (const float*)d_in[16];
    const float* Wl    = (const float*)d_in[17];
    const float* bl    = (const float*)d_in[18];

    const int N = in_sizes[0] / 16;     // 100000
    const int E = in_sizes[2];          // 1600000
    const int C = in_sizes[18];         // 51
    const int G = out_size / C;         // 256
    const int* src  = ei;
    const int* dsti = ei + E;

    // Workspace carve-up (256B aligned)
    char* ws = (char*)d_ws;
    size_t ofs = 0;
    auto carve = [&](size_t bytes) -> void* {
        void* p = ws + ofs;
        ofs = (ofs + bytes + 255) & ~(size_t)255;
        return p;
    };
    int*   cnt = (int*)  carve((size_t)N * 4 * sizeof(int));
    float* hA  = (float*)carve((size_t)N * 64 * sizeof(float));
    float* hB  = (float*)carve((size_t)N * 64 * sizeof(float));
    float* xr  = (float*)carve((size_t)4 * N * 64 * sizeof(float));
    float* g   = (float*)carve((size_t)G * 64 * sizeof(float));
    float* g1  = (float*)carve((size_t)G * 128 * sizeof(float));
    float* g2  = (float*)carve((size_t)G * 256 * sizeof(float));
    (void)ws_size; (void)n_in;

    // --- per-(dst,rel) edge counts (shared by all layers) ---
    fill_i32<<<(N * 4 + 255) / 256, 256, 0, stream>>>(cnt, 0, N * 4);
    count_edges<<<(E + 255) / 256, 256, 0, stream>>>(dsti, et, cnt, E);

    const int MT = N / 16;  // 6250 row tiles

    // ================= Layer 1: 16 -> 16 =================
    gemm16_wmma_f32<<<dim3(MT, 1, 4), 32, 0, stream>>>(x, W1, nullptr, xr, N, 16, 16, 16 * 16, N * 16);
    gemm16_wmma_f32<<<dim3(MT, 1, 1), 32, 0, stream>>>(x, root1, b1, hA, N, 16, 16, 0, 0);
    {
        long long total = (long long)E * 16;
        scatter_mean<<<(unsigned)((total + 255) / 256), 256, 0, stream>>>(
            xr, src, dsti, et, cnt, hA, total, 16, N, 4);
    }
    relu_inplace<<<(N * 16 + 255) / 256, 256, 0, stream>>>(hA, N * 16);

    // ================= Layer 2: 16 -> 32 =================
    gemm16_wmma_f32<<<dim3(MT, 2, 4), 32, 0, stream>>>(hA, W2, nullptr, xr, N, 16, 32, 16 * 32, N * 32);
    gemm16_wmma_f32<<<dim3(MT, 2, 1), 32, 0, stream>>>(hA, root2, b2, hB, N, 16, 32, 0, 0);
    {
        long long total = (long long)E * 32;
        scatter_mean<<<(unsigned)((total + 255) / 256), 256, 0, stream>>>(
            xr, src, dsti, et, cnt, hB, total, 32, N, 5);
    }
    relu_inplace<<<(N * 32 + 255) / 256, 256, 0, stream>>>(hB, N * 32);

    // ================= Layer 3: 32 -> 64 (no relu) =================
    gemm16_wmma_f32<<<dim3(MT, 4, 4), 32, 0, stream>>>(hB, W3, nullptr, xr, N, 32, 64, 32 * 64, N * 64);
    gemm16_wmma_f32<<<dim3(MT, 4, 1), 32, 0, stream>>>(hB, root3, b3, hA, N, 32, 64, 0, 0);
    {
        long long total = (long long)E * 64;
        scatter_mean<<<(unsigned)((total + 255) / 256), 256, 0, stream>>>(
            xr, src, dsti, et, cnt, hA, total, 64, N, 6);
    }

    // ================= Pool: segment max over batch =================
    fill_f32<<<(G * 64 + 255) / 256, 256, 0, stream>>>(g, -INFINITY, G * 64);
    pool_max<<<(N * 64 + 255) / 256, 256, 0, stream>>>(hA, batch, g, N);

    // ================= MLP head + softmax =================
    linear_act<<<G, 128, 0, stream>>>(g, Wf1, bf1, g1, 64, 128, 1);
    linear_act<<<G, 256, 0, stream>>>(g1, Wh, bh, g2, 128, 256, 1);
    final_linear_softmax<<<G, 64, 0, stream>>>(g2, Wl, bl, (float*)d_out, 256, C);
}